// ChamferSilhouetteLoss_17154099381030
// MI455X (gfx1250) — compile-verified
//
#include <hip/hip_runtime.h>
#include <hip/hip_bf16.h>

// CDNA5 (gfx1250) Chamfer loss, WMMA-bound formulation:
//   - prologue packs each point as {z0, z1, z2, ||z||^2-or-BIG} (mask baked in)
//   - hot loop per 16x16 tile: one b64 load (B operand direct from memory),
//     one V_WMMA_F32_16X16X4_F32 with SRC2 = ||x||^2 (so D = full distance),
//     eight running-min updates. Clamp deferred past the min (they commute).
//   - wave32 shuffle min-reductions, deterministic final sum (no float atomics)

typedef __attribute__((ext_vector_type(2))) float v2f;
typedef __attribute__((ext_vector_type(8))) float v8f;

#define NBATCH 4
#define NP1 8192
#define NP2 8192
#define BIGF 1e10f
#define TILES_PER_SIDE (NP2 / 16)     // 512
#define WAVES_PER_BLOCK 8
#define BLOCKS_X (TILES_PER_SIDE / WAVES_PER_BLOCK)  // 64
#define NPARTIALS (2 * NBATCH * TILES_PER_SIDE)      // 4096

// workspace layout (bytes)
#define WS_PARTIALS 0
#define WS_PPRED    (NPARTIALS * 4)                        // 16 KB in
#define WS_PTARG    (WS_PPRED + NBATCH * NP1 * 4 * 4)      // +512 KB
// total: 16 KB + 512 KB + 512 KB = ~1.04 MB

// ---- Pack points into WMMA-B-ready records: {z0, z1, z2, ||z||^2 or BIG}.
__global__ __launch_bounds__(256)
void pack_points(const float* __restrict__ pts, const int* __restrict__ lens,
                 float4* __restrict__ out, int npts, int mask)
{
    const int idx = blockIdx.x * 256 + threadIdx.x;   // over NBATCH*npts
    if (idx >= NBATCH * npts) return;
    const int b = idx / npts;
    const int j = idx - b * npts;
    const float z0 = pts[(size_t)idx * 3 + 0];
    const float z1 = pts[(size_t)idx * 3 + 1];
    const float z2 = pts[(size_t)idx * 3 + 2];
    float y2 = fmaf(z2, z2, fmaf(z1, z1, z0 * z0));
    if (mask && j >= lens[b]) y2 = BIGF;              // padded target -> huge distance
    out[idx] = make_float4(z0, z1, z2, y2);           // global_store_b128
}

__global__ __launch_bounds__(256)
void chamfer_wmma_pass(const float* __restrict__ pred,
                       const float* __restrict__ targ,
                       const int*   __restrict__ lens,
                       const float* __restrict__ packedPred,
                       const float* __restrict__ packedTarg,
                       float*       __restrict__ partials)
{
    const int lane = threadIdx.x & 31;
    const int wave = threadIdx.x >> 5;      // 0..7
    const int m    = lane & 15;             // row/col slot within 16
    const int hi   = lane >> 4;             // 0 -> K=0,1 / rows 0-7; 1 -> K=2,3 / rows 8-15

    const int b     = blockIdx.y;           // batch
    const int pass  = blockIdx.z;           // 0: x->y, 1: y->x
    const int qtile = blockIdx.x * WAVES_PER_BLOCK + wave;  // 0..511
    const int Ly    = lens[b];

    const float* qpts;                      // raw query coords (for A + ||x||^2)
    const float* rpk;                       // packed reference records (for B)
    int nref, qlimit;
    if (pass == 0) {                        // queries = pred, refs = target (masked)
        qpts = pred + (size_t)b * NP1 * 3;
        rpk  = packedTarg + (size_t)b * NP2 * 4;
        nref = NP2; qlimit = NP1;
    } else {                                // queries = target (rows masked), refs = pred
        qpts = targ + (size_t)b * NP2 * 3;
        rpk  = packedPred + (size_t)b * NP1 * 4;
        nref = NP1; qlimit = Ly;
    }

    // ---- A operand (16x4 f32): row = (-2x0, -2x1, -2x2, 1.0); the 1.0 picks up
    //      the ||y||^2 slot of B, so the WMMA emits -2 x.y + ||y||^2.
    const int qi = qtile * 16 + m;
    const float q0 = qpts[qi * 3 + 0];
    const float q1 = qpts[qi * 3 + 1];
    const float q2 = qpts[qi * 3 + 2];
    const float x2 = fmaf(q2, q2, fmaf(q1, q1, q0 * q0));

    v2f A;
    A.x = hi ? (-2.0f * q2) : (-2.0f * q0); // K=2 (hi) or K=0
    A.y = hi ? 1.0f         : (-2.0f * q1); // K=3 = 1.0 (hi) or K=1

    // ---- SRC2 accumulator = ||x||^2 per C/D element (row r+8*hi), loop-invariant:
    //      D[r] = ||x||^2 + ||y||^2 - 2 x.y  straight out of the matrix unit.
    v8f Cin;
#pragma unroll
    for (int r = 0; r < 8; ++r)
        Cin[r] = __shfl(x2, r + 8 * hi, 32);  // lanes 0..15 carry rows 0..15

    float rowMin[8];
#pragma unroll
    for (int r = 0; r < 8; ++r) rowMin[r] = BIGF;

    // ---- Sweep reference tiles: b64 load + WMMA + 8 mins per 16x16 tile.
    const int ntiles = nref / 16;
    for (int jt = 0; jt < ntiles; ++jt) {
        const int j = jt * 16 + m;
        // B 4x16: lanes 0-15 take (z0,z1) = K0,1; lanes 16-31 take (z2,y2m) = K2,3.
        const v2f B = *(const v2f*)(rpk + (size_t)j * 4 + hi * 2);

        const v8f D = __builtin_amdgcn_wmma_f32_16x16x4_f32(
                /*neg_a=*/false, A, /*neg_b=*/false, B,
                /*c_mod=*/(short)0, Cin, /*reuse_a=*/false, /*reuse_b=*/false);

#pragma unroll
        for (int r = 0; r < 8; ++r)
            rowMin[r] = fminf(rowMin[r], D[r]);   // clamp deferred (min/max commute)

        // keep the streaming packed-reference read warm (L2-resident)
        __builtin_prefetch((const void*)(rpk + (size_t)(j + 128) * 4 + hi * 2), 0, 3);
    }

    // ---- Row-min across the 16 lanes of each half (columns), then sum valid rows.
    float acc = 0.0f;
#pragma unroll
    for (int r = 0; r < 8; ++r) {
        float v = rowMin[r];
        v = fminf(v, __shfl_xor(v, 1, 32));
        v = fminf(v, __shfl_xor(v, 2, 32));
        v = fminf(v, __shfl_xor(v, 4, 32));
        v = fminf(v, __shfl_xor(v, 8, 32));   // all 16 lanes of the half hold the row min
        v = fmaxf(v, 0.0f);                   // deferred clamp (matches ref's max(d2,0))
        const int row = qtile * 16 + r + 8 * hi;
        if (row < qlimit) acc += v;           // pass2: drop padded target rows
    }
    acc += __shfl_xor(acc, 16, 32);           // combine halves (rows 0-7 + rows 8-15)

    if (lane == 0) {
        const float denom = (pass == 0) ? (float)NP1 : (float)Ly;
        const float scale = 1.0f / (denom * (float)NBATCH);
        const int gid = (pass * NBATCH + b) * TILES_PER_SIDE + qtile;
        partials[gid] = acc * scale;
    }
}

__global__ __launch_bounds__(256)
void chamfer_reduce(const float* __restrict__ partials, float* __restrict__ out)
{
    __shared__ float sm[256];
    float s = 0.0f;
    for (int i = threadIdx.x; i < NPARTIALS; i += 256)  // fixed order -> deterministic
        s += partials[i];
    sm[threadIdx.x] = s;
    __syncthreads();
    for (int off = 128; off > 0; off >>= 1) {
        if ((int)threadIdx.x < off) sm[threadIdx.x] += sm[threadIdx.x + off];
        __syncthreads();
    }
    if (threadIdx.x == 0) out[0] = sm[0];
}

extern "C" void kernel_launch(void* const* d_in, const int* in_sizes, int n_in,
                              void* d_out, int out_size, void* d_ws, size_t ws_size,
                              hipStream_t stream)
{
    (void)in_sizes; (void)n_in; (void)out_size; (void)ws_size;

    const float* pred = (const float*)d_in[0];   // [N,P1,3] f32
    const float* targ = (const float*)d_in[1];   // [N,P2,3] f32
    const int*   lens = (const int*)d_in[2];     // [N] i32
    // d_in[3] = num_neighbours (==1) -- K=1 hardcoded as pure min.

    char* ws = (char*)d_ws;
    float*  partials   = (float*)(ws + WS_PARTIALS);
    float*  packedPred = (float*)(ws + WS_PPRED);
    float*  packedTarg = (float*)(ws + WS_PTARG);

    float* out = (float*)d_out;                  // scalar f32 loss

    // 1) pack both point sets (mask baked into target records)
    const int npk = (NBATCH * NP1 + 255) / 256;  // 128 blocks
    pack_points<<<npk, 256, 0, stream>>>(pred, lens, (float4*)packedPred, NP1, 0);
    pack_points<<<npk, 256, 0, stream>>>(targ, lens, (float4*)packedTarg, NP2, 1);

    // 2) both chamfer passes
    dim3 grid(BLOCKS_X, NBATCH, 2);              // 64 x 4 x 2 blocks, 8 waves each
    chamfer_wmma_pass<<<grid, 256, 0, stream>>>(pred, targ, lens,
                                                packedPred, packedTarg, partials);

    // 3) deterministic scalar reduction
    chamfer_reduce<<<1, 256, 0, stream>>>(partials, out);
}